// VM2M0711_1949915152905
// MI455X (gfx1250) — compile-verified
//
#include <hip/hip_runtime.h>
#include <hip/hip_bf16.h>

typedef __attribute__((ext_vector_type(2))) float v2f;
typedef __attribute__((ext_vector_type(8))) float v8f;

// ---------------------------------------------------------------------------
// Kernel 1: pixel_shuffle + 3x3 conv (WMMA f32 implicit GEMM) + BN + ReLU +
//           per-image dynamic 32->4->1 MLP + sigmoid  ->  z0 scratch plane
// grid (8, 8, 32)  block 256 (8 waves)
// LDS: halo tile 40.5KB + full 9-tap B-layout weights 36KB + 1KB misc ~ 77.5KB
// (gfx1250 allows up to 320KB LDS per workgroup)
// ---------------------------------------------------------------------------
__global__ __launch_bounds__(256) void fused_head_kernel(
    const float* __restrict__ emb,      // (32, 512, 32, 32)
    const float* __restrict__ inck,     // (32, 137)
    const float* __restrict__ conv_w,   // (32, 32, 3, 3)
    const float* __restrict__ conv_b,   // (32)
    const float* __restrict__ bn_g,
    const float* __restrict__ bn_b,
    const float* __restrict__ bn_m,
    const float* __restrict__ bn_v,
    float* __restrict__ z0)             // (32, 128, 128) scratch
{
    __shared__ float smem[10368 + 9216 + 256];
    float* inLDS  = smem;            // [y*18+x][c]  18*18*32 halo tile
    float* xLDS   = smem;            // alias after GEMM: [pix][c] 256*32
    float* wLDS   = smem + 10368;    // [t][c][o] 9*32*32 (all taps, B layout)
    float* kLDS   = smem + 19584;    // 137 dynamic-kernel floats
    float* scLDS  = smem + 19728;    // 32 BN scale
    float* offLDS = smem + 19760;    // 32 BN offset (conv bias folded)

    const int tid = threadIdx.x;
    const int n   = blockIdx.z;
    const int HY  = blockIdx.y * 16;
    const int WX  = blockIdx.x * 16;

    // ---- stage pixel-shuffled halo tile (zero pad at image borders) -------
    const float* embn = emb + (size_t)n * 512 * 32 * 32;
    for (int i = tid; i < 18 * 18 * 32; i += 256) {
        const int c   = i & 31;
        const int pix = i >> 5;
        const int y   = pix / 18;
        const int x   = pix - y * 18;
        const int H   = HY + y - 1;
        const int W   = WX + x - 1;
        float v = 0.f;
        if ((unsigned)H < 128u && (unsigned)W < 128u) {
            const int cc = c * 16 + (H & 3) * 4 + (W & 3);   // pixel-shuffle
            v = embn[(cc * 32 + (H >> 2)) * 32 + (W >> 2)];
        }
        inLDS[i] = v;
    }
    // ---- stage ALL conv weights once, walking global memory linearly -----
    for (int i = tid; i < 9216; i += 256) {        // i = (o*32 + c)*9 + t
        const int o   = i / 288;
        const int rem = i - o * 288;
        const int c   = rem / 9;
        const int t   = rem - c * 9;
        wLDS[(t * 32 + c) * 32 + o] = conv_w[i];
    }
    if (tid < 137) kLDS[tid] = inck[n * 137 + tid];
    if (tid < 32) {
        const float sc = bn_g[tid] * rsqrtf(bn_v[tid] + 1e-5f);
        scLDS[tid]  = sc;
        offLDS[tid] = (conv_b[tid] - bn_m[tid]) * sc + bn_b[tid];
    }
    __syncthreads();

    const int wave = tid >> 5;
    const int lane = tid & 31;
    const int half = lane >> 4;      // 0: lanes 0-15, 1: lanes 16-31
    const int ml   = lane & 15;
    const int nb   = (wave >> 2) << 4;   // out-channel base for this wave

    v8f acc[4] = {};                 // 4 wave-tiles of 16 pixels x 16 och

    // ---- implicit-GEMM: straight 288-WMMA K loop (9 taps x 8 k-steps) ----
    for (int t = 0; t < 9; ++t) {
        const int dy = t / 3, dx = t - dy * 3;
        const float* bcol = &wLDS[(t * 32) * 32 + nb + ml];
#pragma unroll
        for (int kk = 0; kk < 8; ++kk) {
            const int c0 = kk * 4 + 2 * half;   // per-lane K start
            v2f bv; bv.x = bcol[c0 * 32]; bv.y = bcol[(c0 + 1) * 32];
#pragma unroll
            for (int j = 0; j < 4; ++j) {
                const int py = ((wave & 3) << 2) + j;   // tile row 0..15
                const float2 a2 = *(const float2*)
                    (&inLDS[((py + dy) * 18 + dx + ml) * 32] + c0);
                v2f av; av.x = a2.x; av.y = a2.y;
                acc[j] = __builtin_amdgcn_wmma_f32_16x16x4_f32(
                    false, av, false, bv, (short)0, acc[j], false, false);
            }
        }
    }

    __syncthreads();   // all GEMM reads of inLDS complete -> alias as xLDS

    // ---- epilogue: bias+BN+ReLU from C-fragment layout -> xLDS[pix][c] ---
    {
        const int och  = nb + ml;
        const float sc = scLDS[och];
        const float of = offLDS[och];
#pragma unroll
        for (int j = 0; j < 4; ++j) {
            const int py = ((wave & 3) << 2) + j;
#pragma unroll
            for (int r = 0; r < 8; ++r) {
                const int m = r + 8 * half;         // VGPR r: M=r / M=r+8
                xLDS[(py * 16 + m) * 32 + och] = fmaxf(acc[j][r] * sc + of, 0.f);
            }
        }
    }
    __syncthreads();

    // ---- per-pixel dynamic MLP (w1 4x32, w2 1x4) + sigmoid ---------------
    {
        const int p = tid;
        float h0 = kLDS[132], h1 = kLDS[133], h2 = kLDS[134], h3 = kLDS[135];
        const float* xp = &xLDS[p * 32];
#pragma unroll
        for (int c = 0; c < 32; ++c) {
            const float xc = xp[c];
            h0 = fmaf(kLDS[c],      xc, h0);
            h1 = fmaf(kLDS[32 + c], xc, h1);
            h2 = fmaf(kLDS[64 + c], xc, h2);
            h3 = fmaf(kLDS[96 + c], xc, h3);
        }
        h0 = fmaxf(h0, 0.f); h1 = fmaxf(h1, 0.f);
        h2 = fmaxf(h2, 0.f); h3 = fmaxf(h3, 0.f);
        float y = kLDS[136];
        y = fmaf(kLDS[128], h0, y);
        y = fmaf(kLDS[129], h1, y);
        y = fmaf(kLDS[130], h2, y);
        y = fmaf(kLDS[131], h3, y);
        const float s = 1.f / (1.f + __expf(-y));
        const int H = HY + (p >> 4), W = WX + (p & 15);
        z0[((size_t)n * 128 + H) * 128 + W] = s;
    }
}

// ---------------------------------------------------------------------------
// Kernel 2: 5x5 2-channel smooth conv over [sigmoid plane, coarse mask]
// ---------------------------------------------------------------------------
__global__ __launch_bounds__(256) void smooth_kernel(
    const float* __restrict__ z0,     // (32, 128, 128)
    const float* __restrict__ mask,   // (32, 128, 128)
    const float* __restrict__ sw,     // (1, 2, 5, 5)
    const float* __restrict__ sb,     // (1)
    float* __restrict__ out)          // (32, 1, 128, 128)
{
    __shared__ float w[50];
    __shared__ float bias;
    if (threadIdx.x < 50) w[threadIdx.x] = sw[threadIdx.x];
    if (threadIdx.x == 0) bias = sb[0];
    __syncthreads();

    const int idx = blockIdx.x * 256 + threadIdx.x;
    const int n   = idx >> 14;
    const int rem = idx & 16383;
    const int H   = rem >> 7, W = rem & 127;
    const float* zp = z0   + (size_t)n * 16384;
    const float* mp = mask + (size_t)n * 16384;

    float acc = bias;
#pragma unroll
    for (int dy = 0; dy < 5; ++dy) {
        const int h = H + dy - 2;
        if ((unsigned)h >= 128u) continue;
#pragma unroll
        for (int dx = 0; dx < 5; ++dx) {
            const int x = W + dx - 2;
            if ((unsigned)x >= 128u) continue;
            const int off = h * 128 + x;
            acc = fmaf(w[dy * 5 + dx],      zp[off], acc);
            acc = fmaf(w[25 + dy * 5 + dx], mp[off], acc);
        }
    }
    out[idx] = acc;
}

// ---------------------------------------------------------------------------
extern "C" void kernel_launch(void* const* d_in, const int* in_sizes, int n_in,
                              void* d_out, int out_size, void* d_ws, size_t ws_size,
                              hipStream_t stream) {
    const float* emb    = (const float*)d_in[0];   // embedding
    const float* inck   = (const float*)d_in[1];   // inc_kernels
    const float* masks  = (const float*)d_in[2];   // coarse_masks
    const float* conv_w = (const float*)d_in[3];
    const float* conv_b = (const float*)d_in[4];
    const float* bn_g   = (const float*)d_in[5];
    const float* bn_b   = (const float*)d_in[6];
    const float* bn_m   = (const float*)d_in[7];
    const float* bn_v   = (const float*)d_in[8];
    const float* sw     = (const float*)d_in[9];
    const float* sb     = (const float*)d_in[10];
    float* out = (float*)d_out;
    float* z0  = (float*)d_ws;                     // 32*128*128*4 = 2 MB

    dim3 g1(8, 8, 32);
    fused_head_kernel<<<g1, 256, 0, stream>>>(emb, inck, conv_w, conv_b,
                                              bn_g, bn_b, bn_m, bn_v, z0);
    smooth_kernel<<<2048, 256, 0, stream>>>(z0, masks, sw, sb, out);
}